// MoeLayer_72722386255907
// MI455X (gfx1250) — compile-verified
//
#include <hip/hip_runtime.h>
#include <hip/hip_bf16.h>
#include <math.h>

// ---------------- problem constants (match reference) ----------------
#define NUM_EXPERTS 8
#define TOP_K 2
#define N_EMBD 1024
#define N_TOKENS (4 * 2048)          // B*T = 8192

// ---------------- tiling ----------------
#define BUCKET_CAP N_TOKENS          // worst case: every token to one expert
#define TILE_M 32                    // tokens per block tile (2 wmma row-tiles)
#define TILE_N 128                   // output dims per block tile (4 waves x 32)
#define CHUNK_K 32                   // K staged in LDS per pipeline stage
#define LDS_PAD 4                    // row stride 36 dwords: 16B-aligned rows,
                                     // bank-conflict-free strided reads
#define LDS_STRIDE (CHUNK_K + LDS_PAD)
#define MTILES_PER_EXPERT (BUCKET_CAP / TILE_M)   // 256

typedef __attribute__((ext_vector_type(2))) float v2f;
typedef __attribute__((ext_vector_type(4))) float v4f;
typedef __attribute__((ext_vector_type(8))) float v8f;
typedef __attribute__((ext_vector_type(4))) int   v4i;

// ---------------- CDNA5 async global->LDS path (probe-guarded) -------
#if __has_builtin(__builtin_amdgcn_global_load_async_to_lds_b128)
#define HAVE_ASYNC_LDS 1
typedef __attribute__((address_space(1))) v4i as1_v4i;   // global int4*
typedef __attribute__((address_space(3))) v4i as3_v4i;   // LDS int4*
#endif

__device__ __forceinline__ void copy16_to_lds(void* lds, const void* gmem) {
#ifdef HAVE_ASYNC_LDS
  // global_load_async_to_lds_b128: per-lane 16B DMA, tracked by ASYNCcnt
  __builtin_amdgcn_global_load_async_to_lds_b128(
      (as1_v4i*)gmem, (as3_v4i*)lds, /*offset=*/0, /*cpol=*/0);
#else
  *(v4f*)lds = *(const v4f*)gmem;
#endif
}

__device__ __forceinline__ void wait_async_copies() {
#ifdef HAVE_ASYNC_LDS
#if __has_builtin(__builtin_amdgcn_s_wait_asynccnt)
  __builtin_amdgcn_s_wait_asynccnt(0);
#else
  asm volatile("s_wait_asynccnt 0x0" ::: "memory");
#endif
#endif
}

// ---------------------------------------------------------------------
// Kernel 1: zero a float region (workspace and output accumulator)
// ---------------------------------------------------------------------
__global__ void zero_f32(float* __restrict__ p, int n) {
  int i = blockIdx.x * blockDim.x + threadIdx.x;
  int stride = gridDim.x * blockDim.x;
  for (; i < n; i += stride) p[i] = 0.0f;
}

// ---------------------------------------------------------------------
// Kernel 2: gating + routing. One wave32 per token.
// ---------------------------------------------------------------------
__global__ void gate_route(const float* __restrict__ x,
                           const float* __restrict__ gate_w,
                           unsigned int* __restrict__ fill,
                           int* __restrict__ bucket_tok,
                           float* __restrict__ bucket_w) {
  const int lane = threadIdx.x & 31;
  const int wave = threadIdx.x >> 5;
  const int n = blockIdx.x * (blockDim.x >> 5) + wave;
  if (n >= N_TOKENS) return;

  float acc[NUM_EXPERTS];
#pragma unroll
  for (int e = 0; e < NUM_EXPERTS; ++e) acc[e] = 0.0f;

  const float* xr = x + (size_t)n * N_EMBD;
  for (int c = lane; c < N_EMBD; c += 32) {
    float xv = xr[c];
#pragma unroll
    for (int e = 0; e < NUM_EXPERTS; ++e)
      acc[e] += xv * gate_w[e * N_EMBD + c];
  }
#pragma unroll
  for (int e = 0; e < NUM_EXPERTS; ++e) {
#pragma unroll
    for (int off = 16; off > 0; off >>= 1)
      acc[e] += __shfl_xor(acc[e], off, 32);
  }

  if (lane == 0) {
    int e0 = 0;
    float l0 = acc[0];
#pragma unroll
    for (int e = 1; e < NUM_EXPERTS; ++e)
      if (acc[e] > l0) { l0 = acc[e]; e0 = e; }
    int e1 = (e0 == 0) ? 1 : 0;
    float l1 = acc[e1];
#pragma unroll
    for (int e = 0; e < NUM_EXPERTS; ++e)
      if (e != e0 && acc[e] > l1) { l1 = acc[e]; e1 = e; }

    float t = __expf(l1 - l0);            // softmax over {l0, l1}, l0 >= l1
    float w0 = 1.0f / (1.0f + t);
    float w1 = 1.0f - w0;

    unsigned p0 = atomicAdd(&fill[e0], 1u);
    bucket_tok[e0 * BUCKET_CAP + p0] = n;
    bucket_w [e0 * BUCKET_CAP + p0] = w0;
    unsigned p1 = atomicAdd(&fill[e1], 1u);
    bucket_tok[e1 * BUCKET_CAP + p1] = n;
    bucket_w [e1 * BUCKET_CAP + p1] = w1;
  }
}

// ---------------------------------------------------------------------
// Kernel 3: bucketed expert GEMM, V_WMMA_F32_16X16X4_F32.
//   Block = 128 threads (4 wave32s). Block tile = 32 tokens x 128 dims.
//   Each wave: 32x32 output block = 2x2 WMMA register tiles (4 x v8f),
//   sharing A/B fragments -> 4 LDS loads per 4 WMMAs.
//   Double-buffered LDS staged by async global->LDS DMA (ASYNCcnt).
// ---------------------------------------------------------------------
__global__ void __launch_bounds__(128)
expert_gemm(const float* __restrict__ x,
            const float* __restrict__ expert_w,
            const unsigned int* __restrict__ fill,
            const int* __restrict__ bucket_tok,
            const float* __restrict__ bucket_w,
            float* __restrict__ out) {
  const int e     = blockIdx.x / MTILES_PER_EXPERT;
  const int mtile = blockIdx.x % MTILES_PER_EXPERT;
  const unsigned count = fill[e];
  if ((unsigned)(mtile * TILE_M) >= count) return;   // block-uniform early exit

  const int col0 = blockIdx.y * TILE_N;
  const int tid  = threadIdx.x;
  const int lane = tid & 31;
  const int wave = tid >> 5;

  __shared__ int   s_tok[TILE_M];
  __shared__ float s_wgt[TILE_M];
  __shared__ __align__(16) float sx[2][TILE_M][LDS_STRIDE];
  __shared__ __align__(16) float sw[2][TILE_N][LDS_STRIDE];

  if (tid < TILE_M) {
    int slot = mtile * TILE_M + tid;
    if ((unsigned)slot < count) {
      s_tok[tid] = bucket_tok[e * BUCKET_CAP + slot];
      s_wgt[tid] = bucket_w [e * BUCKET_CAP + slot];
    } else {                                  // padded tail: contributes 0
      s_tok[tid] = 0;
      s_wgt[tid] = 0.0f;
    }
  }
  __syncthreads();                            // s_tok ready for address gen

  const float* We = expert_w + (size_t)e * N_EMBD * N_EMBD;

  // Stage one CHUNK_K slab into buffer b (per-lane 16B copies, coalesced).
  auto stage = [&](int b, int kc) {
    // x tile: 32 rows x 128B  -> 256 x b128
    for (int j = tid; j < (TILE_M * CHUNK_K) / 4; j += 128) {
      int r = j >> 3, c = (j & 7) << 2;
      copy16_to_lds(&sx[b][r][c], x + (size_t)s_tok[r] * N_EMBD + kc + c);
    }
    // W tile: 128 rows x 128B -> 1024 x b128
    for (int j = tid; j < (TILE_N * CHUNK_K) / 4; j += 128) {
      int r = j >> 3, c = (j & 7) << 2;
      copy16_to_lds(&sw[b][r][c], We + (size_t)(col0 + r) * N_EMBD + kc + c);
    }
  };

  v8f acc00 = {}, acc01 = {}, acc10 = {}, acc11 = {};

  // WMMA lane mapping (ISA 7.12.2):
  //   A 16x4: lane -> row m = lane&15; v2f = {K=kb, K=kb+1}, kb = (lane>>4)*2
  //   B 4x16: lane -> col n = lane&15; same K pairing
  //   C/D:    v8f[r] -> row r (lanes 0-15) / r+8 (lanes 16-31), col lane&15
  const int am  = lane & 15;
  const int kb  = (lane >> 4) << 1;
  const int bn0 = wave * 32 + (lane & 15);
  const int bn1 = bn0 + 16;

  stage(0, 0);
  int cur = 0;
  for (int kc = 0; kc < N_EMBD; kc += CHUNK_K) {
    wait_async_copies();
    __syncthreads();                 // slab `cur` complete for all waves
    if (kc + CHUNK_K < N_EMBD)
      stage(cur ^ 1, kc + CHUNK_K);  // overlap next DMA with compute

#pragma unroll
    for (int k4 = 0; k4 < CHUNK_K; k4 += 4) {
      v2f a0, a1, b0, b1;
      a0.x = sx[cur][am][k4 + kb];       a0.y = sx[cur][am][k4 + kb + 1];
      a1.x = sx[cur][16 + am][k4 + kb];  a1.y = sx[cur][16 + am][k4 + kb + 1];
      b0.x = sw[cur][bn0][k4 + kb];      b0.y = sw[cur][bn0][k4 + kb + 1];
      b1.x = sw[cur][bn1][k4 + kb];      b1.y = sw[cur][bn1][k4 + kb + 1];
      acc00 = __builtin_amdgcn_wmma_f32_16x16x4_f32(false, a0, false, b0,
                                                    (short)0, acc00, false, false);
      acc01 = __builtin_amdgcn_wmma_f32_16x16x4_f32(false, a0, false, b1,
                                                    (short)0, acc01, false, false);
      acc10 = __builtin_amdgcn_wmma_f32_16x16x4_f32(false, a1, false, b0,
                                                    (short)0, acc10, false, false);
      acc11 = __builtin_amdgcn_wmma_f32_16x16x4_f32(false, a1, false, b1,
                                                    (short)0, acc11, false, false);
    }
    cur ^= 1;
    // no trailing barrier: next iteration's wait+barrier fences buffer reuse
  }

  // epilogue: scale rows by routing weight, accumulate into out
  const int nn  = lane & 15;
  const int mhi = (lane >> 4) * 8;
#pragma unroll
  for (int r = 0; r < 8; ++r) {
    int m0 = r + mhi;                  // rows 0..15 tile
    int m1 = 16 + r + mhi;             // rows 16..31 tile
    int d0 = col0 + wave * 32 + nn;    // cols 0..15 tile
    int d1 = d0 + 16;                  // cols 16..31 tile
    atomicAdd(&out[(size_t)s_tok[m0] * N_EMBD + d0], acc00[r] * s_wgt[m0]);
    atomicAdd(&out[(size_t)s_tok[m0] * N_EMBD + d1], acc01[r] * s_wgt[m0]);
    atomicAdd(&out[(size_t)s_tok[m1] * N_EMBD + d0], acc10[r] * s_wgt[m1]);
    atomicAdd(&out[(size_t)s_tok[m1] * N_EMBD + d1], acc11[r] * s_wgt[m1]);
  }
}

// ---------------------------------------------------------------------
// Launch
// ---------------------------------------------------------------------
extern "C" void kernel_launch(void* const* d_in, const int* in_sizes, int n_in,
                              void* d_out, int out_size, void* d_ws, size_t ws_size,
                              hipStream_t stream) {
  (void)in_sizes; (void)n_in; (void)out_size; (void)ws_size;

  const float* x        = (const float*)d_in[0];   // (B,T,C) f32
  const float* gate_w   = (const float*)d_in[1];   // (E,C)   f32
  const float* expert_w = (const float*)d_in[2];   // (E,C,C) f32
  float*       out      = (float*)d_out;           // (B,T,C) f32

  unsigned char* ws = (unsigned char*)d_ws;
  unsigned int* fill       = (unsigned int*)ws;                       // 8 u32
  int*          bucket_tok = (int*)(ws + 1024);                       // E*CAP i32
  float*        bucket_w   = (float*)(ws + 1024 +
                              (size_t)NUM_EXPERTS * BUCKET_CAP * 4);  // E*CAP f32

  const int ws_words = (1024 + 2 * NUM_EXPERTS * BUCKET_CAP * 4) / 4;

  zero_f32<<<256, 256, 0, stream>>>((float*)ws, ws_words);
  zero_f32<<<2048, 256, 0, stream>>>(out, N_TOKENS * N_EMBD);

  gate_route<<<N_TOKENS / 8, 256, 0, stream>>>(x, gate_w, fill, bucket_tok, bucket_w);

  dim3 grid(NUM_EXPERTS * MTILES_PER_EXPERT, N_EMBD / TILE_N);
  expert_gemm<<<grid, 128, 0, stream>>>(x, expert_w, fill, bucket_tok, bucket_w, out);
}